// TransitionDown_32899449487857
// MI455X (gfx1250) — compile-verified
//
#include <hip/hip_runtime.h>

typedef __attribute__((ext_vector_type(2))) float v2f;
typedef __attribute__((ext_vector_type(8))) float v8f;

#define BATCH  8
#define NPTS   8192
#define NCEN   2048
#define KNB    32
#define RADIUS 0.2f
#define FDIM   64

// ---------------------------------------------------------------------------
// Kernel 1: farthest point sampling. One block per batch, 1024 threads,
// points register-resident (8 per thread). 2047 sequential argmax steps,
// only TWO barriers per step:
//   B1: publish per-wave argmax partials (32 entries in LDS)
//   -> every wave redundantly shuffle-reduces the 32 partials (deterministic,
//      so all threads agree on the winner w with no extra barrier); the
//      owning thread writes the winner's coords from registers to LDS
//   B2: publish coords + protect partial slots for next iteration
// ---------------------------------------------------------------------------
__global__ __launch_bounds__(1024)
void fps_kernel(const float* __restrict__ p, float* __restrict__ pce)
{
    const int b = blockIdx.x;
    const int t = threadIdx.x;
    const float* pb = p + (size_t)b * NPTS * 3;

    float x[8], y[8], z[8], md[8];
#pragma unroll
    for (int k = 0; k < 8; ++k) {
        int i = t + (k << 10);
        x[k]  = pb[i * 3 + 0];
        y[k]  = pb[i * 3 + 1];
        z[k]  = pb[i * 3 + 2];
        md[k] = 1e10f;                       // reference init: 1e10
    }

    __shared__ float s_v[32];
    __shared__ int   s_i[32];
    __shared__ float s_c[3];

    // first center is index 0 (reference scan emits incoming 'far' = 0 first)
    float cx = pb[0], cy = pb[1], cz = pb[2];
    if (t == 0) {
        float* o = pce + (size_t)b * NCEN * 3;
        o[0] = cx; o[1] = cy; o[2] = cz;
    }

    const int lane = t & 31;
    const int wid  = t >> 5;

    for (int it = 1; it < NCEN; ++it) {
        float bv = -1.0f; int bi = 0;
#pragma unroll
        for (int k = 0; k < 8; ++k) {
            float dx = x[k] - cx, dy = y[k] - cy, dz = z[k] - cz;
            float d  = dx * dx + dy * dy + dz * dz;
            md[k] = fminf(md[k], d);
            if (md[k] > bv) { bv = md[k]; bi = t + (k << 10); }  // strict > keeps lowest idx
        }
        // wave32 argmax reduction (ties -> lower index, matches jnp.argmax)
        for (int off = 16; off > 0; off >>= 1) {
            float ov = __shfl_down(bv, off);
            int   oi = __shfl_down(bi, off);
            if (ov > bv || (ov == bv && oi < bi)) { bv = ov; bi = oi; }
        }
        if (lane == 0) { s_v[wid] = bv; s_i[wid] = bi; }
        __syncthreads();                                   // B1

        // every wave redundantly reduces the 32 partials (bank-conflict-free)
        bv = s_v[lane]; bi = s_i[lane];
        for (int off = 16; off > 0; off >>= 1) {
            float ov = __shfl_down(bv, off);
            int   oi = __shfl_down(bi, off);
            if (ov > bv || (ov == bv && oi < bi)) { bv = ov; bi = oi; }
        }
        const int w = __shfl(bi, 0);                       // all threads agree
        if ((w & 1023) == t) {                             // owner has coords
            int k = w >> 10;
            s_c[0] = x[k]; s_c[1] = y[k]; s_c[2] = z[k];
        }
        __syncthreads();                                   // B2
        cx = s_c[0]; cy = s_c[1]; cz = s_c[2];
        if (t == 0) {
            float* o = pce + ((size_t)b * NCEN + it) * 3;
            o[0] = cx; o[1] = cy; o[2] = cz;
        }
    }
}

// ---------------------------------------------------------------------------
// Kernel 2: ball query + gather + max-pool. One wave owns 16 centers.
//
// WMMA operand assignment (no LDS transpose needed):
//   A (16x4, per tile)      : row m = point base+m = (px, py, pz, ||p||^2)
//   B (4x16, LOOP-INVARIANT): col n = center n     = (-2cx, -2cy, -2cz, 1)
//   D[m][n] = ||p_m||^2 - 2 c_n . p_m ;  in-ball  <=>  D <= r^2 - ||c_n||^2
//
// D layout: center n's column lives in lanes {n, n+16} (M=0..7 / M=8..15),
// the same lanes that hold center n's threshold. Each lane packs an 8-bit
// in-ball mask in registers; one shfl merges halves; a ctz loop appends hit
// indices in ascending order -> identical to sort(where(sq>r^2,N,idx))[:K]
// with first-index padding. Two WMMAs per iteration = 32 points.
// ---------------------------------------------------------------------------
__global__ __launch_bounds__(256)
void ballquery_gather_max_kernel(const float* __restrict__ p,
                                 const float* __restrict__ f,
                                 const float* __restrict__ pce,
                                 float* __restrict__ fout)
{
    __shared__ int s_nbr[8][16 * KNB];               // per-wave neighbour lists

    const int lane = threadIdx.x & 31;
    const int wid  = threadIdx.x >> 5;
    const int gw   = blockIdx.x * 8 + wid;           // global wave id
    const int b    = gw / (NCEN / 16);
    const int s0   = (gw % (NCEN / 16)) * 16;

    // center (lane & 15), duplicated on both half-waves
    const int cm = lane & 15;
    const float* c = pce + ((size_t)b * NCEN + s0 + cm) * 3;
    const float cx = c[0], cy = c[1], cz = c[2];
    const float cc = cx * cx + cy * cy + cz * cz;
    const float thr = RADIUS * RADIUS - cc;          // valid on both half-waves

    // loop-invariant B matrix (centers): lanes 0-15 K=0,1 ; lanes 16-31 K=2,3
    v2f bcen;
    if (lane < 16) { bcen.x = -2.0f * cx; bcen.y = -2.0f * cy; }
    else           { bcen.x = -2.0f * cz; bcen.y = 1.0f; }

    int cnt = (lane < 16) ? 0 : KNB;                 // upper half: dummy-full

    const float* pb = p + (size_t)b * NPTS * 3;
    const v8f zero = {0.f, 0.f, 0.f, 0.f, 0.f, 0.f, 0.f, 0.f};

    for (int base = 0; base < NPTS; base += 32) {
        // A tiles: point base+cm and base+16+cm (each point read by 2 lanes)
        const float* q0 = pb + (size_t)(base + cm) * 3;
        const float* q1 = q0 + 48;                   // +16 points
        float p0x = q0[0], p0y = q0[1], p0z = q0[2];
        float p1x = q1[0], p1y = q1[1], p1z = q1[2];
        v2f a0, a1;
        if (lane < 16) {
            a0.x = p0x; a0.y = p0y;
            a1.x = p1x; a1.y = p1y;
        } else {
            a0.x = p0z; a0.y = p0x * p0x + p0y * p0y + p0z * p0z;
            a1.x = p1z; a1.y = p1x * p1x + p1y * p1y + p1z * p1z;
        }

        v8f d0 = __builtin_amdgcn_wmma_f32_16x16x4_f32(
            false, a0, false, bcen, (short)0, zero, false, false);
        v8f d1 = __builtin_amdgcn_wmma_f32_16x16x4_f32(
            false, a1, false, bcen, (short)0, zero, false, false);

        // pack 8-bit in-ball masks (lane n: M=0..7 ; lane n+16: M=8..15)
        unsigned mA = 0u, mB = 0u;
#pragma unroll
        for (int r = 0; r < 8; ++r) {
            mA |= (d0[r] <= thr ? 1u : 0u) << r;
            mB |= (d1[r] <= thr ? 1u : 0u) << r;
        }
        // merge upper half into selection lanes (0..15): 32-bit point mask
        unsigned hiA = (unsigned)__shfl((int)mA, lane | 16);
        unsigned hiB = (unsigned)__shfl((int)mB, lane | 16);
        unsigned mask = (lane < 16)
                      ? (mA | (hiA << 8) | (mB << 16) | (hiB << 24)) : 0u;

        // ordered append (ctz walks ascending indices); hits are rare
        while (mask && cnt < KNB) {
            int nbit = __builtin_ctz(mask);
            mask &= mask - 1u;
            s_nbr[wid][lane * KNB + cnt] = base + nbit;
            ++cnt;
        }
        if (__all(cnt >= KNB)) break;                // uniform early exit
    }
    __builtin_amdgcn_wave_barrier();
    asm volatile("" ::: "memory");

    // pad with first in-ball index (ball always contains its own center)
    if (lane < 16) {
        if (cnt == 0) { s_nbr[wid][lane * KNB] = 0; cnt = 1; }
        int first = s_nbr[wid][lane * KNB];
        for (int j = cnt; j < KNB; ++j) s_nbr[wid][lane * KNB + j] = first;
    }
    __builtin_amdgcn_wave_barrier();
    asm volatile("" ::: "memory");

    // gather + max-pool: lane owns channels {lane, lane+32}; rows are 256B
    // coalesced and L2-resident (f = 16 MB << 192 MB L2). int4 index reads
    // let the compiler clause 8 global loads per group.
    for (int m = 0; m < 16; ++m) {
        float m0 = -3.402823466e+38f, m1 = m0;
        const int* nb = &s_nbr[wid][m * KNB];
#pragma unroll
        for (int j = 0; j < KNB; j += 4) {
            int4 nn = *(const int4*)(nb + j);        // uniform -> LDS broadcast
            const float* f0 = f + ((size_t)b * NPTS + nn.x) * FDIM;
            const float* f1 = f + ((size_t)b * NPTS + nn.y) * FDIM;
            const float* f2 = f + ((size_t)b * NPTS + nn.z) * FDIM;
            const float* f3 = f + ((size_t)b * NPTS + nn.w) * FDIM;
            m0 = fmaxf(m0, f0[lane]);      m1 = fmaxf(m1, f0[lane + 32]);
            m0 = fmaxf(m0, f1[lane]);      m1 = fmaxf(m1, f1[lane + 32]);
            m0 = fmaxf(m0, f2[lane]);      m1 = fmaxf(m1, f2[lane + 32]);
            m0 = fmaxf(m0, f3[lane]);      m1 = fmaxf(m1, f3[lane + 32]);
        }
        float* o = fout + ((size_t)b * NCEN + s0 + m) * FDIM;
        o[lane]      = m0;
        o[lane + 32] = m1;
    }
}

// ---------------------------------------------------------------------------
extern "C" void kernel_launch(void* const* d_in, const int* in_sizes, int n_in,
                              void* d_out, int out_size, void* d_ws, size_t ws_size,
                              hipStream_t stream)
{
    const float* f = (const float*)d_in[0];   // (8, 8192, 64)
    const float* p = (const float*)d_in[1];   // (8, 8192, 3)

    float* out = (float*)d_out;
    float* fce = out;                                     // (8, 2048, 64)
    float* pce = out + (size_t)BATCH * NCEN * FDIM;       // (8, 2048, 3)

    hipLaunchKernelGGL(fps_kernel, dim3(BATCH), dim3(1024), 0, stream, p, pce);

    const int nwaves = BATCH * NCEN / 16;                 // 1024 waves
    hipLaunchKernelGGL(ballquery_gather_max_kernel,
                       dim3(nwaves / 8), dim3(256), 0, stream,
                       p, f, pce, fce);
}